// QuantumLikeLinear_43215960932473
// MI455X (gfx1250) — compile-verified
//
#include <hip/hip_runtime.h>
#include <hip/hip_bf16.h>

// Problem constants (from the reference)
#define DIM    1024
#define BATCH  65536
#define TPB    128                 // threads per block == rows per block tile
#define CHUNK  64                  // columns processed per pipeline stage
#define NCHUNK (DIM / CHUNK)       // 16
#define PS     (CHUNK + 4)         // padded LDS row stride (floats), 16B aligned
#define F4PR   (CHUNK / 4)         // float4 per tile row = 16
#define ITERS  ((TPB * F4PR) / TPB) // async-load instructions per wave per chunk = 16

// ---- gfx1250 async global->LDS path (CDNA5 ASYNCcnt engine) ---------------
#if defined(__AMDGCN__) && \
    __has_builtin(__builtin_amdgcn_global_load_async_to_lds_b128)
#define USE_ASYNC 1
typedef int v4i_b128 __attribute__((ext_vector_type(4)));
typedef __attribute__((address_space(1))) v4i_b128* gptr_b128;  // global
typedef __attribute__((address_space(3))) v4i_b128* lptr_b128;  // LDS
#else
#define USE_ASYNC 0
#endif

template <int N>
__device__ __forceinline__ void wait_async_le() {
#if USE_ASYNC
#if __has_builtin(__builtin_amdgcn_s_wait_asynccnt)
  __builtin_amdgcn_s_wait_asynccnt(N);
#else
  asm volatile("s_wait_asynccnt %0" :: "i"(N) : "memory");
#endif
#endif
}

// Streaming Givens-rotation-chain kernel (out = sequential 1024-rotation chain
// applied per row). Per-row first-order scan in registers; global traffic is
// double-buffered async-loaded into LDS so every VMEM access is coalesced and
// HBM latency overlaps compute + stores.
__global__ __launch_bounds__(TPB, 1) void givens_chain_kernel(
    const float* __restrict__ x,
    const float* __restrict__ ang,
    float* __restrict__ out)
{
  __shared__ float Cc[DIM];
  __shared__ float Ss[DIM];
  __shared__ float tile[2][TPB * PS];   // 2 x (128 x 68) floats = 68 KB

  const int tid      = threadIdx.x;
  const int rowBase  = blockIdx.x * TPB;
  const size_t myRow = (size_t)(rowBase + tid);

  // Per-block trig tables (2048 sincos per block; negligible vs 1 MB traffic)
  for (int i = tid; i < DIM; i += TPB) {
    float t = ang[i];
    Cc[i] = cosf(t);
    Ss[i] = sinf(t);
  }

  // Cooperative coalesced tile load: chunk 'ch' of all block rows -> tile[buf].
  auto issue_loads = [&](int ch, int buf) {
    const int c0 = ch * CHUNK;
#pragma unroll
    for (int it = 0; it < ITERS; ++it) {
      const int f  = tid + TPB * it;     // float4 index within tile
      const int r  = f >> 4;             // tile row (F4PR = 16)
      const int c4 = f & (F4PR - 1);     // float4 column
      const float* gp = x + ((size_t)(rowBase + r) * DIM + c0 + c4 * 4);
      float* lp = &tile[buf][r * PS + c4 * 4];
#if USE_ASYNC
      __builtin_amdgcn_global_load_async_to_lds_b128(
          (gptr_b128)gp, (lptr_b128)lp, 0, 0);
#else
      *(float4*)lp = *(const float4*)gp;
#endif
    }
  };

  float a    = 0.0f;   // scan carry a_p
  float out0 = 0.0f;   // rotation-0 result for col 0 (finalized by rot 1023)

  issue_loads(0, 0);   // prime the pipeline

  for (int ch = 0; ch < NCHUNK; ++ch) {
    const int c0  = ch * CHUNK;
    const int buf = ch & 1;

    // Prefetch next chunk into the other buffer, then wait only for the
    // current chunk's loads (loads retire in order on ASYNCcnt).
    float halo = 0.0f;
    if (ch + 1 < NCHUNK) {
      issue_loads(ch + 1, buf ^ 1);
      halo = x[myRow * DIM + c0 + CHUNK];  // first column of next chunk
      wait_async_le<ITERS>();
    } else {
      wait_async_le<0>();
    }
    __syncthreads();

    // ---- per-row rotation chain over this chunk (in-place in LDS) -------
    float* trow = &tile[buf][tid * PS];
    int q = 0;
    if (ch == 0) {
      const float x0 = trow[0];
      const float x1 = trow[1];
      out0 = Cc[0] * x0 - Ss[0] * x1;    // provisional out[0]
      a    = Ss[0] * x0 + Cc[0] * x1;    // a_1
      q = 1;                             // rotation 0 done
    }
    for (; q < CHUNK; ++q) {
      const int p = c0 + q;              // rotation index == output column
      const float xn = (q < CHUNK - 1) ? trow[q + 1]
                     : ((ch < NCHUNK - 1) ? halo : out0); // rot 1023 uses out0
      const float cp = Cc[p], sp = Ss[p];
      const float o  = cp * a - sp * xn; // out[p]
      trow[q] = o;
      a = sp * a + cp * xn;              // a_{p+1}; after p=1023 this IS out[0]
    }
    if (ch == NCHUNK - 1) out[myRow * DIM] = a;  // final wrap-around col 0

    __syncthreads();

    // ---- cooperative coalesced store: LDS tile -> global (fire & forget) --
    // ds_load -> VGPR -> global_store_b128: the VGPR dependency guarantees the
    // LDS reads complete before the barrier, so the buffer is safely reusable
    // without waiting for global store completion.
#pragma unroll
    for (int it = 0; it < ITERS; ++it) {
      const int f  = tid + TPB * it;
      const int r  = f >> 4;
      const int c4 = f & (F4PR - 1);
      const float* lp = &tile[buf][r * PS + c4 * 4];
      float* gp = out + ((size_t)(rowBase + r) * DIM + c0 + c4 * 4);
      if (ch == 0 && c4 == 0) {
        // global col 0 is written only by the owning thread in the last chunk
        gp[1] = lp[1]; gp[2] = lp[2]; gp[3] = lp[3];
      } else {
        *(float4*)gp = *(const float4*)lp;
      }
    }
    __syncthreads();   // buffer 'buf' may now be overwritten (chunk ch+2 loads)
  }
}

extern "C" void kernel_launch(void* const* d_in, const int* in_sizes, int n_in,
                              void* d_out, int out_size, void* d_ws, size_t ws_size,
                              hipStream_t stream) {
  const float* x   = (const float*)d_in[0];   // (65536, 1024) f32
  const float* ang = (const float*)d_in[1];   // (1024,) f32
  float* out = (float*)d_out;                 // (65536, 1024) f32

  dim3 grid(BATCH / TPB);   // 512 blocks of 128 threads (4 waves, wave32)
  givens_chain_kernel<<<grid, TPB, 0, stream>>>(x, ang, out);
}